// SubsetLayer_69870527971643
// MI455X (gfx1250) — compile-verified
//
#include <hip/hip_runtime.h>
#include <hip/hip_bf16.h>

// Problem constants (match the reference).
#define BATCH        256
#define NFEAT        20000
#define TOPK         256
#define NSAMP        8
#define TPB          256
#define CHUNK        79            // ceil(NFEAT / TPB)
#define NFEAT_BULK   (NFEAT & ~1023)   // 19456: b128 bulk region (1024 elems/iter)

// ---------------- CDNA5 async global<->LDS helpers (GVS mode) ----------------
// global_load_async_to_lds_*  : vdst = LDS byte address, vaddr = 32-bit byte
// offset, saddr = 64-bit SGPR base.  Tracked with ASYNCcnt.
__device__ __forceinline__ void async_ld_b32(unsigned ldsaddr, unsigned voff, const void* base) {
    asm volatile("global_load_async_to_lds_b32 %0, %1, %2"
                 :: "v"(ldsaddr), "v"(voff), "s"(base) : "memory");
}
__device__ __forceinline__ void async_ld_b128(unsigned ldsaddr, unsigned voff, const void* base) {
    asm volatile("global_load_async_to_lds_b128 %0, %1, %2"
                 :: "v"(ldsaddr), "v"(voff), "s"(base) : "memory");
}
__device__ __forceinline__ void async_st_b32(unsigned voff, unsigned ldsaddr, const void* base) {
    asm volatile("global_store_async_from_lds_b32 %0, %1, %2"
                 :: "v"(voff), "v"(ldsaddr), "s"(base) : "memory");
}
__device__ __forceinline__ void async_st_b128(unsigned voff, unsigned ldsaddr, const void* base) {
    asm volatile("global_store_async_from_lds_b128 %0, %1, %2"
                 :: "v"(voff), "v"(ldsaddr), "s"(base) : "memory");
}
__device__ __forceinline__ void wait_async0() {
    asm volatile("s_wait_asynccnt 0" ::: "memory");
}
__device__ __forceinline__ void wait_ds0() {
    asm volatile("s_wait_dscnt 0" ::: "memory");
}

// Monotone float->uint key: larger float => larger key.
__device__ __forceinline__ unsigned mkey(unsigned u) {
    return (u & 0x80000000u) ? ~u : (u | 0x80000000u);
}

__global__ __launch_bounds__(TPB)
void subset_topk_mask_kernel(const float* __restrict__ logits, float* __restrict__ out) {
    // 16B-aligned dynamic LDS: required for *_b128 async DMA.
    extern __shared__ __align__(16) unsigned lkeys[];   // NFEAT words = 80000 B
    __shared__ unsigned hist[256];                      // 1024 B (16-mult)
    __shared__ unsigned scanb[256];                     // 1024 B (16-mult)
    __shared__ unsigned s_meta[4];                      // [0]=prefix [1]=want, padded to 16 B

    const int tid = threadIdx.x;
    const int b   = blockIdx.x;
    const float* row = logits + (size_t)b * NFEAT;
    const unsigned lbase = (unsigned)(unsigned long long)(uintptr_t)&lkeys[0];

    // ---- 1) async DMA the row into LDS (coalesced, b128 bulk + b32 tail) ----
    for (int base = 0; base + 1024 <= NFEAT; base += 1024) {
        unsigned off = (unsigned)(base + tid * 4) * 4u;
        async_ld_b128(lbase + off, off, row);
    }
    for (int i = NFEAT_BULK + tid; i < NFEAT; i += TPB) {
        unsigned off = (unsigned)i * 4u;
        async_ld_b32(lbase + off, off, row);
    }
    if (tid == 0) s_meta[1] = TOPK;
    wait_async0();                 // each wave drains its own ASYNCcnt
    __syncthreads();               // then LDS is coherent across the workgroup

    // Contiguous per-thread chunks (index order matters for tie-breaking).
    const int start = tid * CHUNK;
    const int end   = (start + CHUNK < NFEAT) ? (start + CHUNK) : NFEAT;

    // ---- 2) MSB-first 8-bit radix select: find the K-th largest key ----
    unsigned curPrefix = 0;
    for (int p = 3; p >= 0; --p) {
        hist[tid] = 0;
        __syncthreads();
        const unsigned shift = (unsigned)p * 8u;
        const unsigned pmask = (p == 3) ? 0u : (0xFFFFFFFFu << (shift + 8u));
        for (int i = start; i < end; ++i) {
            unsigned key;
            if (p == 3) { key = mkey(lkeys[i]); lkeys[i] = key; }  // transform in place once
            else        { key = lkeys[i]; }
            if ((key & pmask) == curPrefix)
                atomicAdd(&hist[(key >> shift) & 0xFFu], 1u);
        }
        __syncthreads();
        if (tid == 0) {
            unsigned want = s_meta[1], cum = 0;
            int v = 255;
            for (; v > 0; --v) {
                unsigned c = hist[v];
                if (cum + c >= want) break;
                cum += c;
            }
            s_meta[0] = curPrefix | ((unsigned)v << shift);
            s_meta[1] = want - cum;      // rank of target within its bin
        }
        __syncthreads();
        curPrefix = s_meta[0];
    }
    const unsigned thrKey = curPrefix;   // exact K-th largest key
    const unsigned nEqSel = s_meta[1];   // #elements == thrKey to include (lowest index first)

    // ---- 3) rank equal-valued elements by global index (matches top_k ties) ----
    unsigned eq = 0;
    for (int i = start; i < end; ++i) eq += (lkeys[i] == thrKey) ? 1u : 0u;
    scanb[tid] = eq;
    __syncthreads();
    if (tid == 0) {                      // tiny serial exclusive scan over 256 counts
        unsigned run = 0;
        for (int t = 0; t < TPB; ++t) { unsigned c = scanb[t]; scanb[t] = run; run += c; }
    }
    __syncthreads();

    // ---- 4) overwrite LDS with the k-hot mask (as float bits) ----
    unsigned r = scanb[tid];
    for (int i = start; i < end; ++i) {
        unsigned key = lkeys[i];
        float m = 0.0f;
        if (key > thrKey)       m = 1.0f;
        else if (key == thrKey) { if (r < nEqSel) m = 1.0f; ++r; }
        lkeys[i] = __float_as_uint(m);
    }
    __syncthreads();
    wait_ds0();                          // LDS mask fully written before DMA reads it

    // ---- 5) async DMA the mask out 8x (one stream per sample slice) ----
    const size_t rowoff = (size_t)b * NFEAT;
#pragma unroll
    for (int s = 0; s < NSAMP; ++s) {
        float* dst = out + (size_t)s * BATCH * NFEAT + rowoff;
        for (int base = 0; base + 1024 <= NFEAT; base += 1024) {
            unsigned off = (unsigned)(base + tid * 4) * 4u;
            async_st_b128(off, lbase + off, dst);
        }
        for (int i = NFEAT_BULK + tid; i < NFEAT; i += TPB) {
            unsigned off = (unsigned)i * 4u;
            async_st_b32(off, lbase + off, dst);
        }
    }
    wait_async0();
}

extern "C" void kernel_launch(void* const* d_in, const int* in_sizes, int n_in,
                              void* d_out, int out_size, void* d_ws, size_t ws_size,
                              hipStream_t stream) {
    const float* logits = (const float*)d_in[0];   // [B, N, 1] float32
    // d_in[1] = tau, d_in[2] = training: unused in the eval branch.
    float* out = (float*)d_out;                    // [8, B, N, 1] float32

    const size_t shmem = (size_t)NFEAT * sizeof(unsigned);  // 80000 B dynamic LDS
    (void)hipFuncSetAttribute((const void*)subset_topk_mask_kernel,
                              hipFuncAttributeMaxDynamicSharedMemorySize, (int)shmem);

    dim3 grid(BATCH), blk(TPB);
    hipLaunchKernelGGL(subset_topk_mask_kernel, grid, blk, shmem, stream, logits, out);
}